// COTREC_41961830482291
// MI455X (gfx1250) — compile-verified
//
#include <hip/hip_runtime.h>
#include <hip/hip_bf16.h>

// ---------------------------------------------------------------------------
// COTREC forward on gfx1250 (wave32).
//  - SpMM graph conv: float4 gathers + GLOBAL_ATOMIC_ADD_F32 (memory bound).
//  - Attention GEMMs: V_WMMA_F32_16X16X4_F32, one 16x16 tile per wave.
//    B tile staged in LDS (zero-padded, conflict-free strides), unconditional
//    ds_load_b64/global_load_b64 in the inner loop, epilogues fused.
// ---------------------------------------------------------------------------

#define N_NODE 100000
#define EMB    100
#define NNZ    1600000
#define BATCH  512
#define SEQL   50

typedef float v2f __attribute__((ext_vector_type(2)));
typedef float v8f __attribute__((ext_vector_type(8)));

__device__ __forceinline__ void atomic_add_f32(float* p, float v) {
    __hip_atomic_fetch_add(p, v, __ATOMIC_RELAXED, __HIP_MEMORY_SCOPE_AGENT);
}

__device__ __forceinline__ v8f wmma_f32(v2f a, v2f b, v8f c) {
    return __builtin_amdgcn_wmma_f32_16x16x4_f32(false, a, false, b,
                                                 (short)0, c, false, false);
}

// ---------------- SpMM: dst[row] += val * src[col], one thread per (edge, 4-dim chunk)
__global__ void cotrec_spmm(const int* __restrict__ rows, const int* __restrict__ cols,
                            const float* __restrict__ vals, const float* __restrict__ src,
                            float* __restrict__ dst) {
    int idx = blockIdx.x * blockDim.x + threadIdx.x;   // < NNZ*25 = 40M
    if (idx >= NNZ * 25) return;
    int e = idx / 25;
    int q = (idx % 25) * 4;
    int r = rows[e];
    int c = cols[e];
    float v = vals[e];
    const float4 x = *reinterpret_cast<const float4*>(src + (long)c * EMB + q);
    float* d = dst + (long)r * EMB + q;
    atomic_add_f32(d + 0, v * x.x);
    atomic_add_f32(d + 1, v * x.y);
    atomic_add_f32(d + 2, v * x.z);
    atomic_add_f32(d + 3, v * x.w);
}

// ---------------- item = (h0 + h1 + h2) / 3    (float4, n/4 threads)
__global__ void cotrec_combine(const float* __restrict__ h0, const float* __restrict__ h1,
                               const float* __restrict__ h2, float* __restrict__ item, int n4) {
    int i = blockIdx.x * blockDim.x + threadIdx.x;
    if (i >= n4) return;
    const float4 a = ((const float4*)h0)[i];
    const float4 b = ((const float4*)h1)[i];
    const float4 c = ((const float4*)h2)[i];
    float4 o;
    o.x = (a.x + b.x + c.x) * (1.0f / 3.0f);
    o.y = (a.y + b.y + c.y) * (1.0f / 3.0f);
    o.z = (a.z + b.z + c.z) * (1.0f / 3.0f);
    o.w = (a.w + b.w + c.w) * (1.0f / 3.0f);
    ((float4*)item)[i] = o;
}

// ---------------- seq_h gather (padded lookup) + hs = mean over L.  One wave per session.
__global__ void cotrec_gather(const int* __restrict__ sess, const float* __restrict__ slen,
                              const float* __restrict__ item, float* __restrict__ seq,
                              float* __restrict__ hs) {
    int b = blockIdx.x;
    int q = threadIdx.x;                 // lanes 0..24 handle float4 chunks
    if (q >= EMB / 4) return;
    float4 acc = {0.0f, 0.0f, 0.0f, 0.0f};
    for (int l = 0; l < SEQL; ++l) {
        int idx = sess[b * SEQL + l];    // uniform across the wave -> no divergence
        float4 v = {0.0f, 0.0f, 0.0f, 0.0f};
        if (idx != 0) v = *(const float4*)(item + (long)(idx - 1) * EMB + q * 4);
        *(float4*)(seq + ((long)b * SEQL + l) * EMB + q * 4) = v;
        acc.x += v.x; acc.y += v.y; acc.z += v.z; acc.w += v.w;
    }
    const float inv = 1.0f / slen[b];
    float4 o = {acc.x * inv, acc.y * inv, acc.z * inv, acc.w * inv};
    *(float4*)(hs + b * EMB + q * 4) = o;
}

// ---------------- hsb[b,j] = glu1_b[j] + sum_e hs[b,e] * glu2_w[j,e]
__global__ void cotrec_hsglu(const float* __restrict__ hs, const float* __restrict__ glu2_w,
                             const float* __restrict__ glu1_b, float* __restrict__ hsb) {
    int t = blockIdx.x * blockDim.x + threadIdx.x;  // b*EMB + j
    if (t >= BATCH * EMB) return;
    int b = t / EMB, j = t % EMB;
    const float4* hrow = (const float4*)(hs + b * EMB);
    const float4* wrow = (const float4*)(glu2_w + j * EMB);
    float acc = glu1_b[j];
    #pragma unroll 5
    for (int e = 0; e < EMB / 4; ++e) {
        float4 h4 = hrow[e];
        float4 w4 = wrow[e];
        acc += h4.x * w4.x + h4.y * w4.y + h4.z * w4.z + h4.w * w4.w;
    }
    hsb[t] = acc;
}

// ---------------- GEMM1: nh1 = tanh( [pos | seq_h] @ w_1 )   (25600 x 200) @ (200 x 100)
// Block = 8 waves sharing one 16-column B tile (staged in LDS, zero-padded cols>=100).
// A layout (f32 16x4): lanes 0-15 hold {K=k0,k0+1} of row M=lane; lanes 16-31 hold {K=k0+2,k0+3}.
// B layout symmetric with lane = column.  D[:,n] depends only on B[:,n], so padded
// columns never contaminate valid ones -> no guards inside the K loop.
#define BSTRIDE1 202   // 200 + pad: 10*lr mod 64 distinct for lr in [0,16) -> conflict-free
__global__ void cotrec_gemm1(const float* __restrict__ seq, const float* __restrict__ pos,
                             const float* __restrict__ w1, float* __restrict__ nh1) {
    __shared__ __align__(16) float Bt[16 * BSTRIDE1];
    const int tid = threadIdx.x;
    const int n0 = (blockIdx.x / 200) * 16;          // n-tile shared by whole block

    // stage B: Bt[j*BSTRIDE1 + k] = (n0+j < 100) ? w1[k,n0+j] : 0   (coalesced in j)
    for (int t = tid; t < 16 * 200; t += 256) {
        int j = t & 15, k = t >> 4;
        int n = n0 + j;
        Bt[j * BSTRIDE1 + k] = (n < EMB) ? w1[k * EMB + n] : 0.0f;
    }
    __syncthreads();

    const int wave = tid >> 5;
    const int lane = tid & 31;
    const int m0 = ((blockIdx.x % 200) * 8 + wave) * 16;
    const int half = lane >> 4, lr = lane & 15;
    const int m = m0 + lr;                           // A row for this lane
    const float* __restrict__ Apos = pos + (m % SEQL) * EMB + 2 * half;
    const float* __restrict__ Aseq = seq + (long)m * EMB + 2 * half;
    const float* __restrict__ Brow = Bt + lr * BSTRIDE1 + 2 * half;

    v8f acc = {};
    #pragma unroll 5
    for (int k0 = 0; k0 < EMB; k0 += 4) {            // K = 0..99 : pos half
        v2f a = *(const v2f*)(Apos + k0);
        v2f b = *(const v2f*)(Brow + k0);
        acc = wmma_f32(a, b, acc);
    }
    #pragma unroll 5
    for (int k0 = 0; k0 < EMB; k0 += 4) {            // K = 100..199 : seq half
        v2f a = *(const v2f*)(Aseq + k0);
        v2f b = *(const v2f*)(Brow + EMB + k0);
        acc = wmma_f32(a, b, acc);
    }

    const int n = n0 + lr;
    if (n < EMB) {
        #pragma unroll
        for (int r = 0; r < 8; ++r) {
            int mm = m0 + r + 8 * half;              // D: lanes 0-15 -> M=r, 16-31 -> M=r+8
            nh1[(long)mm * EMB + n] = tanhf(acc[r]);
        }
    }
}

// ---------------- GEMM2: nh2 = sigmoid(nh1 @ glu1_w^T + hsb); beta[m] += sum_n nh2*w2[n]
#define BSTRIDE2 102   // 100 + pad: 38*lr mod 64 distinct for lr in [0,16) -> conflict-free
__global__ void cotrec_gemm2(const float* __restrict__ nh1, const float* __restrict__ glu1_w,
                             const float* __restrict__ hsb, const float* __restrict__ w2,
                             float* __restrict__ beta) {
    __shared__ __align__(16) float Bt[16 * BSTRIDE2];
    const int tid = threadIdx.x;
    const int n0 = (blockIdx.x / 200) * 16;

    // stage B: Bt[j*BSTRIDE2 + k] = (n0+j < 100) ? glu1_w[n0+j, k] : 0  (coalesced in k)
    for (int t = tid; t < 16 * EMB; t += 256) {
        int j = t / EMB, k = t % EMB;
        int n = n0 + j;
        Bt[j * BSTRIDE2 + k] = (n < EMB) ? glu1_w[n * EMB + k] : 0.0f;
    }
    __syncthreads();

    const int wave = tid >> 5;
    const int lane = tid & 31;
    const int m0 = ((blockIdx.x % 200) * 8 + wave) * 16;
    const int half = lane >> 4, lr = lane & 15;
    const float* __restrict__ Arow = nh1 + (long)(m0 + lr) * EMB + 2 * half;
    const float* __restrict__ Brow = Bt + lr * BSTRIDE2 + 2 * half;

    v8f acc = {};
    #pragma unroll 5
    for (int k0 = 0; k0 < EMB; k0 += 4) {
        v2f a = *(const v2f*)(Arow + k0);
        v2f b = *(const v2f*)(Brow + k0);
        acc = wmma_f32(a, b, acc);
    }

    const int n = n0 + lr;
    const bool nv = (n < EMB);
    const int ncl = nv ? n : (EMB - 1);              // clamped index: unconditional loads
    float w2n = w2[ncl];
    if (!nv) w2n = 0.0f;                             // kills padded-column contributions
    #pragma unroll
    for (int r = 0; r < 8; ++r) {
        int mm = m0 + r + 8 * half;
        float add = hsb[(mm / SEQL) * EMB + ncl];
        float s = 1.0f / (1.0f + __expf(-(acc[r] + add)));
        float contrib = s * w2n;
        // reduce across the 16-lane half (xor 1,2,4,8 never crosses bit 4)
        contrib += __shfl_xor(contrib, 1);
        contrib += __shfl_xor(contrib, 2);
        contrib += __shfl_xor(contrib, 4);
        contrib += __shfl_xor(contrib, 8);
        if (lr == 0) atomic_add_f32(&beta[mm], contrib);
    }
}

// ---------------- select[b,:] = sum_l beta[b,l]*mask[b,l]*seq_h[b,l,:]   (one wave per b)
__global__ void cotrec_select(const float* __restrict__ seq, const float* __restrict__ beta,
                              const float* __restrict__ mask, float* __restrict__ out) {
    int b = blockIdx.x;
    int q = threadIdx.x;
    if (q >= EMB / 4) return;
    float4 acc = {0.0f, 0.0f, 0.0f, 0.0f};
    for (int l = 0; l < SEQL; ++l) {
        float w = beta[b * SEQL + l] * mask[b * SEQL + l];
        float4 v = *(const float4*)(seq + ((long)b * SEQL + l) * EMB + q * 4);
        acc.x += w * v.x; acc.y += w * v.y; acc.z += w * v.z; acc.w += w * v.w;
    }
    *(float4*)(out + b * EMB + q * 4) = acc;
}

extern "C" void kernel_launch(void* const* d_in, const int* in_sizes, int n_in,
                              void* d_out, int out_size, void* d_ws, size_t ws_size,
                              hipStream_t stream) {
    const int*   adj_rows = (const int*)  d_in[0];
    const int*   adj_cols = (const int*)  d_in[1];
    const float* adj_vals = (const float*)d_in[2];
    const float* embed    = (const float*)d_in[3];
    const int*   sess     = (const int*)  d_in[4];
    const float* slen     = (const float*)d_in[5];
    const float* mask     = (const float*)d_in[6];
    const float* pos      = (const float*)d_in[7];
    const float* w1       = (const float*)d_in[8];
    const float* w2       = (const float*)d_in[9];
    const float* glu1_w   = (const float*)d_in[10];
    const float* glu1_b   = (const float*)d_in[11];
    const float* glu2_w   = (const float*)d_in[12];
    float* out = (float*)d_out;

    const long NE = (long)N_NODE * EMB;              // 10,000,000 floats
    float* ws   = (float*)d_ws;
    float* h1   = ws;                                // [0,      10M)
    float* h2   = ws + NE;                           // [10M,    20M)
    float* item = ws + 2 * NE;                       // [20M,    30M)
    // graph phase done -> reuse [0, 10M):
    float* seqh = ws;                                // 2,560,000
    float* nh1  = ws + 2560000;                      // 2,560,000
    float* hs   = ws + 5120000;                      // 51,200
    float* hsb  = ws + 5171200;                      // 51,200
    float* beta = ws + 5222400;                      // 25,600

    // ---- graph convolution (2 layers) ----
    hipMemsetAsync(h1, 0, NE * sizeof(float), stream);
    hipMemsetAsync(h2, 0, NE * sizeof(float), stream);
    const int spmm_blocks = (NNZ * 25 + 255) / 256;  // 156250
    cotrec_spmm<<<spmm_blocks, 256, 0, stream>>>(adj_rows, adj_cols, adj_vals, embed, h1);
    cotrec_spmm<<<spmm_blocks, 256, 0, stream>>>(adj_rows, adj_cols, adj_vals, h1, h2);
    cotrec_combine<<<(int)((NE / 4 + 255) / 256), 256, 0, stream>>>(embed, h1, h2, item,
                                                                    (int)(NE / 4));

    // ---- session gather + means ----
    cotrec_gather<<<BATCH, 32, 0, stream>>>(sess, slen, item, seqh, hs);
    cotrec_hsglu<<<(BATCH * EMB + 255) / 256, 256, 0, stream>>>(hs, glu2_w, glu1_b, hsb);

    // ---- attention GEMMs (WMMA f32) ----
    hipMemsetAsync(beta, 0, BATCH * SEQL * sizeof(float), stream);
    cotrec_gemm1<<<1400, 256, 0, stream>>>(seqh, pos, w1, nh1);   // 7 n-tiles x 200 blocks
    cotrec_gemm2<<<1400, 256, 0, stream>>>(nh1, glu1_w, hsb, w2, beta);

    // ---- weighted select ----
    cotrec_select<<<BATCH, 32, 0, stream>>>(seqh, beta, mask, out);
    (void)in_sizes; (void)n_in; (void)out_size; (void)ws_size;
}